// GGL_21345987461373
// MI455X (gfx1250) — compile-verified
//
#include <hip/hip_runtime.h>
#include <hip/hip_bf16.h>
#include <float.h>

typedef __attribute__((ext_vector_type(2))) float v2f;
typedef __attribute__((ext_vector_type(8))) float v8f;

#define KTOP 20
#define PADK 12   // K=10 padded to 12 -> 3 x WMMA_F32_16X16X4_F32
#define SSTRIDE 132  // 16x128 score stripe row stride (dwords), bank-safe

// ---------------------------------------------------------------------------
// Kernel 1: atrr[i,p] = sigmoid(x[i,:] @ W[:,p] + b[p]), padded to stride 12.
// Also zero-initializes the per-row max accumulator (scores are > 0).
// ---------------------------------------------------------------------------
__global__ void attr_kernel(const float* __restrict__ x, const float* __restrict__ W,
                            const float* __restrict__ b, float* __restrict__ atrrP,
                            unsigned int* __restrict__ maxbits, int N, int D) {
  int i = blockIdx.x * blockDim.x + threadIdx.x;
  if (i >= N) return;
  float acc[10];
#pragma unroll
  for (int p = 0; p < 10; ++p) acc[p] = b[p];
  const float* xi = x + (size_t)i * D;
  for (int d = 0; d < D; ++d) {
    float xv = xi[d];
    const float* wd = W + (size_t)d * 10;
#pragma unroll
    for (int p = 0; p < 10; ++p) acc[p] = fmaf(xv, wd[p], acc[p]);
  }
  float* o = atrrP + (size_t)i * PADK;
#pragma unroll
  for (int p = 0; p < 10; ++p) o[p] = 1.0f / (1.0f + __expf(-acc[p]));
  o[10] = 0.0f;
  o[11] = 0.0f;
  maxbits[i] = 0u;  // +0.0f bits; all similarities are strictly positive
}

// ---------------------------------------------------------------------------
// Kernel 2: maxval[i] = max_j dot(atrr[i], atrr[j]) via fp32 WMMA 16x16x4.
// One block per 16-row tile; 8 waves split the column tiles.
// C layout: VGPR r, lanes 0-15 -> (M=r, N=lane); lanes 16-31 -> (M=r+8).
// ---------------------------------------------------------------------------
__global__ void __launch_bounds__(256) rowmax_kernel(const float* __restrict__ atrrP,
                                                     unsigned int* __restrict__ maxbits,
                                                     int N) {
  const int lane = threadIdx.x & 31;
  const int wave = threadIdx.x >> 5;
  const int nwaves = blockDim.x >> 5;
  const int half = lane >> 4;
  const int nrow16 = lane & 15;
  const int rt = blockIdx.x;
  const int ntiles = N >> 4;

  v2f afrag[3];
  const size_t arow = (size_t)(rt * 16 + nrow16) * PADK;
#pragma unroll
  for (int kk = 0; kk < 3; ++kk) {
    int k = kk * 4 + 2 * half;
    afrag[kk].x = atrrP[arow + k];
    afrag[kk].y = atrrP[arow + k + 1];
  }

  v8f mx;
#pragma unroll
  for (int r = 0; r < 8; ++r) mx[r] = -FLT_MAX;

  for (int jt = wave; jt < ntiles; jt += nwaves) {
    const size_t bcol = (size_t)(jt * 16 + nrow16) * PADK;
    v2f bfrag[3];
#pragma unroll
    for (int kk = 0; kk < 3; ++kk) {
      int k = kk * 4 + 2 * half;
      bfrag[kk].x = atrrP[bcol + k];
      bfrag[kk].y = atrrP[bcol + k + 1];
    }
    v8f c = {};
#pragma unroll
    for (int kk = 0; kk < 3; ++kk)
      c = __builtin_amdgcn_wmma_f32_16x16x4_f32(false, afrag[kk], false, bfrag[kk],
                                                (short)0, c, false, false);
#pragma unroll
    for (int r = 0; r < 8; ++r) mx[r] = fmaxf(mx[r], c[r]);
  }

  // Row max over N: reduce the 16 lanes within each half (masks 1..8 stay inside).
#pragma unroll
  for (int m = 8; m >= 1; m >>= 1) {
#pragma unroll
    for (int r = 0; r < 8; ++r) mx[r] = fmaxf(mx[r], __shfl_xor(mx[r], m, 32));
  }

  if (nrow16 == 0) {
    int rbase = rt * 16 + half * 8;
#pragma unroll
    for (int r = 0; r < 8; ++r)
      atomicMax((int*)&maxbits[rbase + r], __float_as_int(mx[r]));
  }
}

// ---------------------------------------------------------------------------
// Kernel 3 (fused): WMMA score tiles -> LDS stripe -> per-thread top-20 scan
// -> 16-way sorted-list merge per row. Block owns a 16-row tile.
// ---------------------------------------------------------------------------
__global__ void __launch_bounds__(256) topk_fused_kernel(
    const float* __restrict__ atrrP, const unsigned int* __restrict__ maxbits,
    float* __restrict__ out, int N) {
  const int tid = threadIdx.x;
  const int lane = tid & 31;
  const int wave = tid >> 5;
  const int half = lane >> 4;
  const int nrow16 = lane & 15;
  const int rt = blockIdx.x;
  const int ntiles = N >> 4;        // column tiles
  const int nchunks = ntiles >> 3;  // 8 waves -> 128 cols per chunk

  __shared__ float sscore[16][SSTRIDE];        // 16 rows x 128 cols (+pad)
  __shared__ float mval[KTOP * 256];           // [t][ssub][srow] layout
  __shared__ int midx[KTOP * 256];

  // Producer: A fragments for this block's 16 rows (held in registers).
  v2f afrag[3];
  const size_t arow = (size_t)(rt * 16 + nrow16) * PADK;
#pragma unroll
  for (int kk = 0; kk < 3; ++kk) {
    int k = kk * 4 + 2 * half;
    afrag[kk].x = atrrP[arow + k];
    afrag[kk].y = atrrP[arow + k + 1];
  }

  // Scanner identity: row = lane&15, sublist = tid>>4 (cols ≡ ssub mod 16).
  const int srow = lane & 15;
  const int ssub = tid >> 4;

  float val[KTOP];
  int idx[KTOP];
#pragma unroll
  for (int t = 0; t < KTOP; ++t) { val[t] = -FLT_MAX; idx[t] = 0; }

  for (int c = 0; c < nchunks; ++c) {
    if (c) __syncthreads();  // previous chunk fully scanned

    // ---- produce: wave's 16x16 tile of A_norm ----
    const int jt = c * 8 + wave;
    const int colbase = jt * 16;
    const size_t bcol = (size_t)(colbase + nrow16) * PADK;
    v2f bfrag[3];
#pragma unroll
    for (int kk = 0; kk < 3; ++kk) {
      int k = kk * 4 + 2 * half;
      bfrag[kk].x = atrrP[bcol + k];
      bfrag[kk].y = atrrP[bcol + k + 1];
    }
    v8f acc = {};
#pragma unroll
    for (int kk = 0; kk < 3; ++kk)
      acc = __builtin_amdgcn_wmma_f32_16x16x4_f32(false, afrag[kk], false, bfrag[kk],
                                                  (short)0, acc, false, false);
    // column normalization: this lane's column is colbase + nrow16 (both halves)
    const float inv = 1.0f / __int_as_float((int)maxbits[colbase + nrow16]);
#pragma unroll
    for (int r = 0; r < 8; ++r)
      sscore[r + 8 * half][wave * 16 + nrow16] = acc[r] * inv;

    __syncthreads();

    // ---- scan: 8 columns of this thread's row (j ascending for stability) ----
    const int jb = c * 128;
#pragma unroll
    for (int k = 0; k < 8; ++k) {
      const int col = ssub + 16 * k;
      float s = sscore[srow][col];
      int j = jb + col;
      if (s > val[KTOP - 1]) {  // rare: full sorted insert (stable: strict >)
        float cv = s;
        int ci = j;
#pragma unroll
        for (int t = 0; t < KTOP; ++t) {
          bool cb = cv > val[t];
          float tv = val[t]; int ti = idx[t];
          val[t] = cb ? cv : tv; idx[t] = cb ? ci : ti;
          cv = cb ? tv : cv;     ci = cb ? ti : ci;
        }
      }
    }
  }

  // ---- publish sorted lists: index = t*256 + ssub*16 + srow (bank-safe) ----
#pragma unroll
  for (int t = 0; t < KTOP; ++t) {
    mval[t * 256 + ssub * 16 + srow] = val[t];
    midx[t * 256 + ssub * 16 + srow] = idx[t];
  }
  __syncthreads();

  // ---- merge 16 sorted lists per row; half-wave (wave, half) owns one row ----
  const int mrow = wave * 2 + half;  // 0..15
  const int m = lane & 15;           // this lane merges list (mrow, m)
  const size_t NK = (size_t)N * KTOP;
  int head = 0;
  for (int t = 0; t < KTOP; ++t) {
    float bv = mval[head * 256 + m * 16 + mrow];
    int bj = midx[head * 256 + m * 16 + mrow];
    int bm = m;
#pragma unroll
    for (int msk = 8; msk >= 1; msk >>= 1) {  // argmax within 16-lane half
      float ov = __shfl_xor(bv, msk, 32);
      int oj = __shfl_xor(bj, msk, 32);
      int om = __shfl_xor(bm, msk, 32);
      if (ov > bv || (ov == bv && oj < bj)) { bv = ov; bj = oj; bm = om; }
    }
    if (m == bm) head++;  // winner advances its sorted list
    if (m == 0) {
      const size_t row = (size_t)rt * 16 + mrow;
      out[row * KTOP + t] = bv;                  // values
      out[NK + row * KTOP + t] = (float)row;     // edge_index[0] (rows)
      out[2 * NK + row * KTOP + t] = (float)bj;  // edge_index[1] (cols)
    }
  }
}

// ---------------------------------------------------------------------------
extern "C" void kernel_launch(void* const* d_in, const int* in_sizes, int n_in,
                              void* d_out, int out_size, void* d_ws, size_t ws_size,
                              hipStream_t stream) {
  const float* x = (const float*)d_in[0];
  const float* W = (const float*)d_in[1];
  const float* b = (const float*)d_in[2];
  float* out = (float*)d_out;

  const int P = in_sizes[2];      // 10
  const int D = in_sizes[1] / P;  // 256
  const int N = in_sizes[0] / D;  // 8192

  float* atrrP = (float*)d_ws;                                        // N*12 floats
  unsigned int* maxbits = (unsigned int*)(atrrP + (size_t)N * PADK);  // N u32

  attr_kernel<<<(N + 255) / 256, 256, 0, stream>>>(x, W, b, atrrP, maxbits, N, D);
  rowmax_kernel<<<N / 16, 256, 0, stream>>>(atrrP, maxbits, N);
  topk_fused_kernel<<<N / 16, 256, 0, stream>>>(atrrP, maxbits, out, N);
}